// MessagePassing_84215718740469
// MI455X (gfx1250) — compile-verified
//
#include <hip/hip_runtime.h>
#include <stdint.h>

#define D_FEAT 32

// -------- zero the output (must run every call: atomics accumulate) --------
__global__ __launch_bounds__(256) void zero_kernel(float* __restrict__ out, int n) {
    int i = (blockIdx.x * blockDim.x + threadIdx.x) * 4;
    if (i + 3 < n) {
        *(float4*)(out + i) = make_float4(0.f, 0.f, 0.f, 0.f);
    } else {
        for (; i < n; ++i) out[i] = 0.f;
    }
}

__device__ __forceinline__ unsigned lds_byte_off(const void* p) {
    // Generic pointer to LDS on gfx1250: addr[31:0] is the LDS byte offset.
    return (unsigned)(uintptr_t)p;
}

// -------- edge-parallel scatter-add: one wave32 per 32-edge tile ----------
// lane = feature index (D_FEAT == 32 == wave size), so every x-row gather and
// every atomic update is one fully coalesced 128B wave transaction.
__global__ __launch_bounds__(256) void scatter_add_kernel(
    const float* __restrict__ x,
    const int*   __restrict__ edge_src,
    const int*   __restrict__ edge_dst,
    float*       __restrict__ out,
    int nE) {

    __shared__ int s_src[256];
    __shared__ int s_dst[256];

    const int lane = threadIdx.x & 31;
    const int wave = threadIdx.x >> 5;
    const long long tileBase = ((long long)blockIdx.x * 8 + wave) * 32;
    const long long e = tileBase + lane;

    // Stage this wave's 32 (src,dst) index pairs into LDS via the CDNA5
    // async-to-LDS path (ASYNCcnt; no VGPR round trip, acts as a prefetch).
    if (e < nE) {
        unsigned lsrc = lds_byte_off(&s_src[threadIdx.x]);
        unsigned ldst = lds_byte_off(&s_dst[threadIdx.x]);
        unsigned goff = (unsigned)(e * 4);  // byte offset into index arrays
        asm volatile("global_load_async_to_lds_b32 %0, %1, %2"
                     :: "v"(lsrc), "v"(goff), "s"(edge_src) : "memory");
        asm volatile("global_load_async_to_lds_b32 %0, %1, %2"
                     :: "v"(ldst), "v"(goff), "s"(edge_dst) : "memory");
    }
    asm volatile("s_wait_asynccnt 0" ::: "memory");

    long long rem = (long long)nE - tileBase;
    if (rem <= 0) return;
    int cnt = rem < 32 ? (int)rem : 32;

    // Each lane holds one edge's indices; broadcast per edge via v_readlane
    // (lane select is wave-uniform) -> indices land in SGPRs, so the gather
    // address is scalar-base + lane*4 (saddr-form global_load_b32).
    int my_s = s_src[threadIdx.x];
    int my_d = s_dst[threadIdx.x];

    if (cnt == 32) {
        // Phase 1: 32 independent coalesced gathers (pipeline, one loadcnt wait).
        float    v[32];
        unsigned obyte[32];
#pragma unroll
        for (int i = 0; i < 32; ++i) {
            int s = __builtin_amdgcn_readlane(my_s, i);
            int d = __builtin_amdgcn_readlane(my_d, i);
            v[i]     = x[(long long)s * D_FEAT + lane];           // 128B wave read, L2-resident
            obyte[i] = ((unsigned)d * D_FEAT + lane) * 4u;        // out is 12.8MB: 32-bit offset ok
        }
        // Phase 2: 32 back-to-back device-scope fp32 atomics (STOREcnt, no waits).
#pragma unroll
        for (int i = 0; i < 32; ++i) {
            asm volatile("global_atomic_add_f32 %0, %1, %2 scope:SCOPE_DEV"
                         :: "v"(obyte[i]), "v"(v[i]), "s"(out));
        }
    } else {
        for (int i = 0; i < cnt; ++i) {
            int s = __builtin_amdgcn_readlane(my_s, i);  // i is wave-uniform
            int d = __builtin_amdgcn_readlane(my_d, i);
            float v = x[(long long)s * D_FEAT + lane];
            unsigned ob = ((unsigned)d * D_FEAT + lane) * 4u;
            asm volatile("global_atomic_add_f32 %0, %1, %2 scope:SCOPE_DEV"
                         :: "v"(ob), "v"(v), "s"(out));
        }
    }
}

extern "C" void kernel_launch(void* const* d_in, const int* in_sizes, int n_in,
                              void* d_out, int out_size, void* d_ws, size_t ws_size,
                              hipStream_t stream) {
    const float* x  = (const float*)d_in[0];
    const int*   ei = (const int*)d_in[1];   // edge_index, shape (2, E) flat
    const int nE = in_sizes[1] / 2;
    const int* esrc = ei;
    const int* edst = ei + nE;
    float* out = (float*)d_out;

    // zero the accumulator
    int zthreads = (out_size + 3) / 4;
    zero_kernel<<<(zthreads + 255) / 256, 256, 0, stream>>>(out, out_size);

    // one wave per 32-edge tile, 8 waves (256 threads) per block
    long long waves  = ((long long)nE + 31) / 32;
    int       blocks = (int)((waves + 7) / 8);
    scatter_add_kernel<<<blocks, 256, 0, stream>>>(x, esrc, edst, out, nE);
}